// SAGELayer_49675591746181
// MI455X (gfx1250) — compile-verified
//
#include <hip/hip_runtime.h>
#include <hip/hip_bf16.h>

#define N_NODES 100000
#define N_EDGES 1250000
#define D 64

typedef __attribute__((ext_vector_type(2))) float v2f;
typedef __attribute__((ext_vector_type(8))) float v8f;

// ---------------------------------------------------------------------------
// Kernel 1: zero the workspace accumulators (sum[N*64] f32, cnt[N] f32)
// ---------------------------------------------------------------------------
__global__ void sage_zero_kernel(float* __restrict__ ws, int n) {
    int i = blockIdx.x * blockDim.x + threadIdx.x;
    if (i < n) ws[i] = 0.0f;
}

// ---------------------------------------------------------------------------
// Kernel 2: edge scatter. 16 lanes per edge, 4 floats per lane (float4 gather
// of the 256B feature row + packed f32 atomics into sum[dst]).
// x and sum both fit in the 192MB L2, so atomics resolve at L2 speed.
// ---------------------------------------------------------------------------
__global__ void sage_scatter_kernel(const long long* __restrict__ ei,
                                    const float* __restrict__ x,
                                    float* __restrict__ sum,
                                    float* __restrict__ cnt) {
    long long t = (long long)blockIdx.x * blockDim.x + threadIdx.x;
    long long e = t >> 4;
    if (e >= N_EDGES) return;
    int q = ((int)t & 15) * 4;

    int src = (int)ei[e];             // edge_index[0][e]
    int dst = (int)ei[N_EDGES + e];   // edge_index[1][e]

    const float4 v = *(const float4*)(x + (size_t)src * D + q);
    float* s = sum + (size_t)dst * D + q;
    atomicAdd(s + 0, v.x);
    atomicAdd(s + 1, v.y);
    atomicAdd(s + 2, v.z);
    atomicAdd(s + 3, v.w);
    if (q == 0) atomicAdd(cnt + dst, 1.0f);
}

// ---------------------------------------------------------------------------
// Kernel 3: out = (sum/max(cnt,1)) @ W_l^T + b_l + x @ W_r^T via
// V_WMMA_F32_16X16X4_F32. One wave = one 16x16 output tile.
// blockDim = 128 (4 waves -> 4 column tiles covering D_OUT=64),
// gridDim = 6250 row tiles (100000 = 6250 * 16, exact; EXEC all-ones).
// ---------------------------------------------------------------------------
__global__ void __launch_bounds__(128)
sage_gemm_kernel(const float* __restrict__ sum, const float* __restrict__ cnt,
                 const float* __restrict__ x,   const float* __restrict__ Wl,
                 const float* __restrict__ bl,  const float* __restrict__ Wr,
                 float* __restrict__ out) {
    const int lane = threadIdx.x & 31;
    const int wave = threadIdx.x >> 5;        // column tile 0..3
    const int m0   = blockIdx.x * 16;         // row (node) tile base
    const int n0   = wave * 16;               // output-feature tile base

    const int r  = lane & 15;                 // A row / B column within tile
    const int hi = lane >> 4;                 // 0 -> K{0,1}, 1 -> K{2,3}

    const int node = m0 + r;
    const int col  = n0 + r;

    const float rinv = 1.0f / fmaxf(cnt[node], 1.0f);

    // per-lane base pointers, pre-offset for the K-pair this half-wave owns
    const float* sRow  = sum + (size_t)node * D + hi * 2;
    const float* xRow  = x   + (size_t)node * D + hi * 2;
    const float* wlRow = Wl  + (size_t)col  * D + hi * 2;  // B[k][n] = W[n*64+k]
    const float* wrRow = Wr  + (size_t)col  * D + hi * 2;

    v8f acc = {};
#pragma unroll
    for (int k = 0; k < D; k += 4) {
        v2f a1 = { sRow[k] * rinv, sRow[k + 1] * rinv };   // mean fragment
        v2f b1 = { wlRow[k],       wlRow[k + 1] };
        acc = __builtin_amdgcn_wmma_f32_16x16x4_f32(
            false, a1, false, b1, (short)0, acc, false, false);

        v2f a2 = { xRow[k],  xRow[k + 1] };                // x fragment
        v2f b2 = { wrRow[k], wrRow[k + 1] };
        acc = __builtin_amdgcn_wmma_f32_16x16x4_f32(
            false, a2, false, b2, (short)0, acc, false, false);
    }

    // D layout: VGPR v -> row (v + 8*hi), col = lane&15 (within tile)
    const float bias = bl[col];
    float* outBase = out + (size_t)(m0 + hi * 8) * D + col;
#pragma unroll
    for (int v = 0; v < 8; ++v) {
        outBase[(size_t)v * D] = acc[v] + bias;
    }
}

// ---------------------------------------------------------------------------
extern "C" void kernel_launch(void* const* d_in, const int* in_sizes, int n_in,
                              void* d_out, int out_size, void* d_ws, size_t ws_size,
                              hipStream_t stream) {
    const float*     x   = (const float*)d_in[0];
    const long long* ei  = (const long long*)d_in[1];   // int64 [2, E]
    const float*     Wl  = (const float*)d_in[2];
    const float*     bl  = (const float*)d_in[3];
    const float*     Wr  = (const float*)d_in[4];
    float*           out = (float*)d_out;

    float* sum = (float*)d_ws;                 // [N_NODES * D]
    float* cnt = sum + (size_t)N_NODES * D;    // [N_NODES]

    // 1) zero accumulators
    {
        int n = N_NODES * D + N_NODES;
        int blocks = (n + 255) / 256;
        sage_zero_kernel<<<blocks, 256, 0, stream>>>(sum, n);
    }
    // 2) edge scatter (mean numerator + degree counts)
    {
        long long threads = (long long)N_EDGES * 16;
        int blocks = (int)((threads + 255) / 256);
        sage_scatter_kernel<<<blocks, 256, 0, stream>>>(ei, x, sum, cnt);
    }
    // 3) fused dual-GEMM + bias via f32 WMMA
    {
        sage_gemm_kernel<<<N_NODES / 16, 128, 0, stream>>>(sum, cnt, x, Wl, bl, Wr, out);
    }
}